// TreeEncoder_GATMini_26216480374790
// MI455X (gfx1250) — compile-verified
//
#include <hip/hip_runtime.h>
#include <hip/hip_bf16.h>

// ---------------- problem constants (from reference) ----------------
#define NN      100000      // nodes
#define EE      1600000     // edges (before self-loops)
#define ET      1700000     // edges + self loops
#define IN_DIM  128
#define HID     16
#define HEADS   8
#define H1DIM   128         // HEADS*HID
#define OUT_DIM 64
#define NG      1000        // graphs

typedef __attribute__((ext_vector_type(2))) float v2f;
typedef __attribute__((ext_vector_type(8))) float v8f;

// ---------------- workspace layout (float offsets) ----------------
// [0)                 h1 / x2          : NN*128 = 12,800,000
// [OFF_ASRC1)         a_src1           : NN*8   =    800,000
// [OFF_ADST1)         a_dst1           : NN*8   =    800,000
// [OFF_P1)            p1               : ET*8   = 13,600,000
//     layer2 reuse of P1 region:
//     h2 @ OFF_P1                      : NN*64  =  6,400,000
//     p2 @ OFF_P1+6,400,000            : ET     =  1,700,000
//     a_src2 @ OFF_P1+8,100,000        : NN     =    100,000
//     a_dst2 @ OFF_P1+8,200,000        : NN     =    100,000
// [OFF_DEN1)          denom1           : NN*8   =    800,000
// [OFF_ACC1)          acc1             : NN*128 = 12,800,000
//     layer2 reuse of ACC1 region:
//     acc2 @ OFF_ACC1                  : NN*64  =  6,400,000
//     denom2 @ OFF_ACC1+6,400,000      : NN     =    100,000
//     pool   @ OFF_ACC1+6,500,000      : NG*64  =     64,000
//     cnt    @ OFF_ACC1+6,564,000      : NG     =      1,000
// total = 41,600,000 floats = 166.4 MB
#define OFF_ASRC1 12800000u
#define OFF_ADST1 13600000u
#define OFF_P1    14400000u
#define OFF_DEN1  28000000u
#define OFF_ACC1  28800000u

// ---------------- utility ----------------
__global__ void zero_kernel(float* __restrict__ p, unsigned n) {
    unsigned i = blockIdx.x * blockDim.x + threadIdx.x;
    if (i < n) p[i] = 0.0f;
}

// ---------------- layer-1 GEMM: h1[N,128] = x[N,128] @ W1[128,128] ----------------
// One block = one 16-row M tile (6250 blocks, N%16==0). 8 waves, wave w owns
// column tile [16w,16w+16). x-tile staged in LDS (8 KB). 32 K-steps of
// v_wmma_f32_16x16x4_f32 per wave.
__global__ __launch_bounds__(256) void gemm1_wmma(const float* __restrict__ x,
                                                  const float* __restrict__ W,
                                                  float* __restrict__ h1) {
    __shared__ float xs[16 * 128];
    const unsigned m0 = blockIdx.x * 16u;
    const float4* xv = (const float4*)(x + (size_t)m0 * 128);
    float4* sv = (float4*)xs;
    for (unsigned i = threadIdx.x; i < 512; i += 256) sv[i] = xv[i];
    __syncthreads();

    const unsigned wave = threadIdx.x >> 5;    // 0..7 -> N tile
    const unsigned lane = threadIdx.x & 31;
    const unsigned ml   = lane & 15;           // A row / B col within tile
    const unsigned kh   = lane >> 4;           // K-half select
    const unsigned col  = wave * 16 + ml;

    v8f c = {};
    #pragma unroll 4
    for (unsigned k0 = 0; k0 < 128; k0 += 4) {
        const unsigned k = k0 + 2 * kh;
        v2f a; a.x = xs[ml * 128 + k];          a.y = xs[ml * 128 + k + 1];
        v2f b; b.x = W[(size_t)k * 128 + col];  b.y = W[(size_t)(k + 1) * 128 + col];
        c = __builtin_amdgcn_wmma_f32_16x16x4_f32(false, a, false, b,
                                                  (short)0, c, false, false);
    }
    #pragma unroll
    for (int r = 0; r < 8; ++r) {
        const unsigned row = m0 + (unsigned)r + 8 * kh;
        h1[(size_t)row * 128 + col] = c[r];
    }
}

// ---------------- layer-2 GEMM: h2[N,64] = x2[N,128] @ W2[128,64] ----------------
__global__ __launch_bounds__(128) void gemm2_wmma(const float* __restrict__ x2,
                                                  const float* __restrict__ W,
                                                  float* __restrict__ h2) {
    __shared__ float xs[16 * 128];
    const unsigned m0 = blockIdx.x * 16u;
    const float4* xv = (const float4*)(x2 + (size_t)m0 * 128);
    float4* sv = (float4*)xs;
    for (unsigned i = threadIdx.x; i < 512; i += 128) sv[i] = xv[i];
    __syncthreads();

    const unsigned wave = threadIdx.x >> 5;    // 0..3 -> N tile
    const unsigned lane = threadIdx.x & 31;
    const unsigned ml   = lane & 15;
    const unsigned kh   = lane >> 4;
    const unsigned col  = wave * 16 + ml;

    v8f c = {};
    #pragma unroll 4
    for (unsigned k0 = 0; k0 < 128; k0 += 4) {
        const unsigned k = k0 + 2 * kh;
        v2f a; a.x = xs[ml * 128 + k];         a.y = xs[ml * 128 + k + 1];
        v2f b; b.x = W[(size_t)k * 64 + col];  b.y = W[(size_t)(k + 1) * 64 + col];
        c = __builtin_amdgcn_wmma_f32_16x16x4_f32(false, a, false, b,
                                                  (short)0, c, false, false);
    }
    #pragma unroll
    for (int r = 0; r < 8; ++r) {
        const unsigned row = m0 + (unsigned)r + 8 * kh;
        h2[(size_t)row * 64 + col] = c[r];
    }
}

// ---------------- attention logits, layer 1: one thread per (node, head) ----------------
__global__ void att1_kernel(const float* __restrict__ h1,
                            const float* __restrict__ att_src,  // [8,16]
                            const float* __restrict__ att_dst,  // [8,16]
                            float* __restrict__ a_src, float* __restrict__ a_dst) {
    unsigned gid = blockIdx.x * blockDim.x + threadIdx.x;
    if (gid >= NN * HEADS) return;
    const unsigned n = gid >> 3, h = gid & 7;
    const float4* hp = (const float4*)(h1 + (size_t)n * 128 + h * 16);
    const float4* as = (const float4*)(att_src + h * 16);
    const float4* ad = (const float4*)(att_dst + h * 16);
    float s = 0.f, d = 0.f;
    #pragma unroll
    for (int q = 0; q < 4; ++q) {
        const float4 v = hp[q];
        const float4 a = as[q];
        const float4 b = ad[q];
        s += v.x * a.x + v.y * a.y + v.z * a.z + v.w * a.w;
        d += v.x * b.x + v.y * b.y + v.z * b.z + v.w * b.w;
    }
    a_src[gid] = s;
    a_dst[gid] = d;
}

// ---------------- edge pass A, layer 1: p = exp(leaky_relu), denom scatter ----------------
// max-shift omitted: it cancels exactly in alpha = p/denom, and logits are O(1).
__global__ void edge1a_kernel(const int* __restrict__ esrc, const int* __restrict__ edst,
                              const float* __restrict__ a_src, const float* __restrict__ a_dst,
                              float* __restrict__ p1, float* __restrict__ denom) {
    unsigned gid = blockIdx.x * blockDim.x + threadIdx.x;
    if (gid >= (unsigned)ET * HEADS) return;
    const unsigned e = gid >> 3, h = gid & 7;
    int s, d;
    if (e < EE) { s = esrc[e]; d = edst[e]; } else { s = d = (int)(e - EE); }
    float v = a_src[(size_t)s * 8 + h] + a_dst[(size_t)d * 8 + h];
    v = v > 0.f ? v : 0.2f * v;
    const float p = __expf(v);
    p1[gid] = p;
    atomicAdd(&denom[(size_t)d * 8 + h], p);
}

// ---------------- edge pass B, layer 1: unnormalized message scatter ----------------
// One thread per (edge, 4-channel group): b128 gather from h1 (L2-resident),
// 4 scalar f32 atomics into acc (no vector f32 atomic exists).
__global__ void edge1b_kernel(const int* __restrict__ esrc, const int* __restrict__ edst,
                              const float* __restrict__ p1, const float* __restrict__ h1,
                              float* __restrict__ acc) {
    const unsigned gid = blockIdx.x * blockDim.x + threadIdx.x;  // < ET*32
    if (gid >= (unsigned)ET * 32u) return;
    const unsigned e = gid >> 5, c0 = (gid & 31) * 4, h = c0 >> 4;
    int s, d;
    if (e < EE) { s = esrc[e]; d = edst[e]; } else { s = d = (int)(e - EE); }
    const float  p  = p1[(size_t)e * 8 + h];
    const float4 hv = *(const float4*)(h1 + (size_t)s * 128 + c0);
    float* a = acc + (size_t)d * 128 + c0;
    atomicAdd(a + 0, p * hv.x);
    atomicAdd(a + 1, p * hv.y);
    atomicAdd(a + 2, p * hv.z);
    atomicAdd(a + 3, p * hv.w);
}

// ---------------- node finalize layer 1: normalize + bias + ELU -> x2 ----------------
__global__ void node1_kernel(const float* __restrict__ acc, const float* __restrict__ denom,
                             const float* __restrict__ bias, float* __restrict__ x2) {
    unsigned gid = blockIdx.x * blockDim.x + threadIdx.x;  // < NN*32
    if (gid >= NN * 32u) return;
    const unsigned n = gid >> 5, c0 = (gid & 31) * 4, h = c0 >> 4;
    const float inv = 1.0f / (denom[(size_t)n * 8 + h] + 1e-16f);
    const float4 av = *(const float4*)(acc + (size_t)n * 128 + c0);
    const float4 bv = *(const float4*)(bias + c0);
    float4 r;
    float v;
    v = av.x * inv + bv.x; r.x = v > 0.f ? v : (__expf(v) - 1.0f);
    v = av.y * inv + bv.y; r.y = v > 0.f ? v : (__expf(v) - 1.0f);
    v = av.z * inv + bv.z; r.z = v > 0.f ? v : (__expf(v) - 1.0f);
    v = av.w * inv + bv.w; r.w = v > 0.f ? v : (__expf(v) - 1.0f);
    *(float4*)(x2 + (size_t)n * 128 + c0) = r;
}

// ---------------- attention logits, layer 2: one thread per node ----------------
__global__ void att2_kernel(const float* __restrict__ h2,
                            const float* __restrict__ att_src,  // [1,64]
                            const float* __restrict__ att_dst,  // [1,64]
                            float* __restrict__ a_src, float* __restrict__ a_dst) {
    unsigned n = blockIdx.x * blockDim.x + threadIdx.x;
    if (n >= NN) return;
    const float4* hp = (const float4*)(h2 + (size_t)n * 64);
    const float4* as = (const float4*)att_src;
    const float4* ad = (const float4*)att_dst;
    float s = 0.f, d = 0.f;
    #pragma unroll
    for (int q = 0; q < 16; ++q) {
        const float4 v = hp[q];
        const float4 a = as[q];
        const float4 b = ad[q];
        s += v.x * a.x + v.y * a.y + v.z * a.z + v.w * a.w;
        d += v.x * b.x + v.y * b.y + v.z * b.z + v.w * b.w;
    }
    a_src[n] = s;
    a_dst[n] = d;
}

// ---------------- edge pass A, layer 2 ----------------
__global__ void edge2a_kernel(const int* __restrict__ esrc, const int* __restrict__ edst,
                              const float* __restrict__ a_src, const float* __restrict__ a_dst,
                              float* __restrict__ p2, float* __restrict__ denom) {
    unsigned e = blockIdx.x * blockDim.x + threadIdx.x;
    if (e >= (unsigned)ET) return;
    int s, d;
    if (e < EE) { s = esrc[e]; d = edst[e]; } else { s = d = (int)(e - EE); }
    float v = a_src[s] + a_dst[d];
    v = v > 0.f ? v : 0.2f * v;
    const float p = __expf(v);
    p2[e] = p;
    atomicAdd(&denom[d], p);
}

// ---------------- edge pass B, layer 2 ----------------
__global__ void edge2b_kernel(const int* __restrict__ esrc, const int* __restrict__ edst,
                              const float* __restrict__ p2, const float* __restrict__ h2,
                              float* __restrict__ acc) {
    const unsigned gid = blockIdx.x * blockDim.x + threadIdx.x;  // < ET*16
    if (gid >= (unsigned)ET * 16u) return;
    const unsigned e = gid >> 4, c0 = (gid & 15) * 4;
    int s, d;
    if (e < EE) { s = esrc[e]; d = edst[e]; } else { s = d = (int)(e - EE); }
    const float  p  = p2[e];
    const float4 hv = *(const float4*)(h2 + (size_t)s * 64 + c0);
    float* a = acc + (size_t)d * 64 + c0;
    atomicAdd(a + 0, p * hv.x);
    atomicAdd(a + 1, p * hv.y);
    atomicAdd(a + 2, p * hv.z);
    atomicAdd(a + 3, p * hv.w);
}

// ---------------- node finalize layer 2 + mean-pool scatter ----------------
__global__ void node2_pool_kernel(const float* __restrict__ acc, const float* __restrict__ denom,
                                  const float* __restrict__ bias, const int* __restrict__ batch,
                                  float* __restrict__ pool, float* __restrict__ cnt) {
    unsigned gid = blockIdx.x * blockDim.x + threadIdx.x;  // < NN*16
    if (gid >= NN * 16u) return;
    const unsigned n = gid >> 4, c0 = (gid & 15) * 4;
    const float inv = 1.0f / (denom[n] + 1e-16f);
    const float4 av = *(const float4*)(acc + (size_t)n * 64 + c0);
    const float4 bv = *(const float4*)(bias + c0);
    const int g = batch[n];
    float* pp = pool + (size_t)g * 64 + c0;
    atomicAdd(pp + 0, av.x * inv + bv.x);
    atomicAdd(pp + 1, av.y * inv + bv.y);
    atomicAdd(pp + 2, av.z * inv + bv.z);
    atomicAdd(pp + 3, av.w * inv + bv.w);
    if (c0 == 0) atomicAdd(&cnt[g], 1.0f);
}

__global__ void pool_final_kernel(const float* __restrict__ pool, const float* __restrict__ cnt,
                                  float* __restrict__ out) {
    unsigned gid = blockIdx.x * blockDim.x + threadIdx.x;
    if (gid >= NG * 64u) return;
    out[gid] = pool[gid] / fmaxf(cnt[gid >> 6], 1.0f);
}

// ---------------- launcher ----------------
extern "C" void kernel_launch(void* const* d_in, const int* in_sizes, int n_in,
                              void* d_out, int out_size, void* d_ws, size_t ws_size,
                              hipStream_t stream) {
    const float* x        = (const float*)d_in[0];
    const int*   eidx     = (const int*)d_in[1];     // [2,E] flat: src then dst
    const int*   batch    = (const int*)d_in[2];
    const float* W1       = (const float*)d_in[3];
    const float* att_src1 = (const float*)d_in[4];
    const float* att_dst1 = (const float*)d_in[5];
    const float* bias1    = (const float*)d_in[6];
    const float* W2       = (const float*)d_in[7];
    const float* att_src2 = (const float*)d_in[8];
    const float* att_dst2 = (const float*)d_in[9];
    const float* bias2    = (const float*)d_in[10];
    float* out = (float*)d_out;

    const int* esrc = eidx;
    const int* edst = eidx + EE;

    float* ws     = (float*)d_ws;
    float* h1     = ws;                      // later reused as x2
    float* x2     = ws;
    float* a_src1 = ws + OFF_ASRC1;
    float* a_dst1 = ws + OFF_ADST1;
    float* p1     = ws + OFF_P1;
    float* denom1 = ws + OFF_DEN1;
    float* acc1   = ws + OFF_ACC1;
    // layer-2 aliases
    float* h2     = ws + OFF_P1;
    float* p2     = ws + OFF_P1 + 6400000u;
    float* a_src2 = ws + OFF_P1 + 8100000u;
    float* a_dst2 = ws + OFF_P1 + 8200000u;
    float* acc2   = ws + OFF_ACC1;
    float* denom2 = ws + OFF_ACC1 + 6400000u;
    float* pool   = ws + OFF_ACC1 + 6500000u;
    float* cnt    = ws + OFF_ACC1 + 6564000u;

    const unsigned B = 256;
    auto blocks = [](unsigned n, unsigned b) { return (n + b - 1) / b; };

    // ---- layer 1 ----
    zero_kernel<<<blocks(NN * 8u + NN * 128u, B), B, 0, stream>>>(denom1, NN * 8u + NN * 128u); // denom1+acc1 contiguous
    gemm1_wmma<<<NN / 16, 256, 0, stream>>>(x, W1, h1);
    att1_kernel<<<blocks(NN * 8u, B), B, 0, stream>>>(h1, att_src1, att_dst1, a_src1, a_dst1);
    edge1a_kernel<<<blocks((unsigned)ET * 8u, B), B, 0, stream>>>(esrc, edst, a_src1, a_dst1, p1, denom1);
    edge1b_kernel<<<blocks((unsigned)ET * 32u, B), B, 0, stream>>>(esrc, edst, p1, h1, acc1);
    node1_kernel<<<blocks(NN * 32u, B), B, 0, stream>>>(acc1, denom1, bias1, x2);

    // ---- layer 2 ----
    zero_kernel<<<blocks(6565000u, B), B, 0, stream>>>(acc2, 6565000u); // acc2+denom2+pool+cnt contiguous
    gemm2_wmma<<<NN / 16, 128, 0, stream>>>(x2, W2, h2);
    att2_kernel<<<blocks(NN, B), B, 0, stream>>>(h2, att_src2, att_dst2, a_src2, a_dst2);
    edge2a_kernel<<<blocks((unsigned)ET, B), B, 0, stream>>>(esrc, edst, a_src2, a_dst2, p2, denom2);
    edge2b_kernel<<<blocks((unsigned)ET * 16u, B), B, 0, stream>>>(esrc, edst, p2, h2, acc2);
    node2_pool_kernel<<<blocks(NN * 16u, B), B, 0, stream>>>(acc2, denom2, bias2, batch, pool, cnt);
    pool_final_kernel<<<blocks(NG * 64u, B), B, 0, stream>>>(pool, cnt, out);
}